// MultiScalePatchExtractor_48481590837808
// MI455X (gfx1250) — compile-verified
//
#include <hip/hip_runtime.h>
#include <hip/hip_bf16.h>

typedef __attribute__((ext_vector_type(2))) float v2f;
typedef __attribute__((ext_vector_type(8))) float v8f;

__device__ __forceinline__ v8f wmma4(v2f a, v2f b, v8f c) {
  // V_WMMA_F32_16X16X4_F32: D = A(16x4,f32) * B(4x16,f32) + C(16x16,f32)
  return __builtin_amdgcn_wmma_f32_16x16x4_f32(false, a, false, b, (short)0, c,
                                               false, false);
}

// One wave (32 lanes) processes a tile of 16 patches through the full MLP.
// 8 waves per 256-thread block, each with a private LDS staging region.
template <int P, int S, int WO, int NP>
__global__ __launch_bounds__(256) void mlp_scale_kernel(
    const float* __restrict__ x,
    const float* __restrict__ w1, const float* __restrict__ b1,
    const float* __restrict__ w2, const float* __restrict__ b2,
    const float* __restrict__ w3,
    float* __restrict__ acc, int B) {
  constexpr int PP = P * P;
  constexpr int KC1 = (PP + 3) / 4;          // K chunks for layer 1
  constexpr int TILES = (NP + 15) / 16;      // 16-patch tiles per image
  constexpr int HS1 = 33;                    // padded row stride, h1 (16x32)
  constexpr int HS2 = 17;                    // padded row stride, h2 (16x16)
  __shared__ float lds[8][16 * HS1 + 16 * HS2];

  const int lane = threadIdx.x & 31;
  const int wv = threadIdx.x >> 5;
  float* h1 = &lds[wv][0];
  float* h2 = &lds[wv][16 * HS1];

  int gtile = blockIdx.x * 8 + wv;           // uniform per wave
  int b = gtile / TILES;
  int t = gtile - b * TILES;
  const bool activeWave = (b < B);
  if (!activeWave) { b = 0; t = 0; }         // keep control flow uniform

  const int m = lane & 15;                   // A-matrix row (patch in tile)
  const int koff = (lane >> 4) << 1;         // K sub-offset: 0 or 2
  const int n0 = lane & 15;                  // B-matrix / D column
  const int mbase = (lane >> 4) << 3;        // D rows: mbase..mbase+7

  // ---------------- Layer 1: (16 x PP) @ (PP x 32), bias + ReLU -----------
  int pidx = t * 16 + m;
  if (pidx >= NP) pidx = NP - 1;             // clamp tail (masked at the end)
  const int oy = pidx / WO;
  const int ox = pidx - oy * WO;
  const float* img = x + (long)b * (224 * 224) + (oy * S) * 224 + (ox * S);

  v8f d0 = {}, d1 = {};
#pragma unroll
  for (int kk = 0; kk < KC1; ++kk) {
    const int k0 = kk * 4 + koff;
    const int k1 = k0 + 1;
    v2f av; av.x = 0.f; av.y = 0.f;
    if (k0 < PP) av.x = img[(k0 / P) * 224 + (k0 % P)];
    if (k1 < PP) av.y = img[(k1 / P) * 224 + (k1 % P)];
    v2f bv0; bv0.x = 0.f; bv0.y = 0.f;       // N tile 0: cols 0..15
    v2f bv1; bv1.x = 0.f; bv1.y = 0.f;       // N tile 1: cols 16..31
    if (k0 < PP) { bv0.x = w1[n0 * PP + k0]; bv1.x = w1[(n0 + 16) * PP + k0]; }
    if (k1 < PP) { bv0.y = w1[n0 * PP + k1]; bv1.y = w1[(n0 + 16) * PP + k1]; }
    d0 = wmma4(av, bv0, d0);
    d1 = wmma4(av, bv1, d1);
  }
  {
    const float bias0 = b1[n0];
    const float bias1 = b1[n0 + 16];
#pragma unroll
    for (int r = 0; r < 8; ++r) {
      h1[(mbase + r) * HS1 + n0]      = fmaxf(d0[r] + bias0, 0.f);
      h1[(mbase + r) * HS1 + n0 + 16] = fmaxf(d1[r] + bias1, 0.f);
    }
  }
  __syncthreads();

  // ---------------- Layer 2: (16 x 32) @ (32 x 16), bias + ReLU -----------
  v8f c2 = {};
#pragma unroll
  for (int kk = 0; kk < 8; ++kk) {
    const int k0 = kk * 4 + koff;
    v2f av; av.x = h1[m * HS1 + k0]; av.y = h1[m * HS1 + k0 + 1];
    v2f bv; bv.x = w2[n0 * 32 + k0]; bv.y = w2[n0 * 32 + k0 + 1];
    c2 = wmma4(av, bv, c2);
  }
  {
    const float bias = b2[n0];
#pragma unroll
    for (int r = 0; r < 8; ++r)
      h2[(mbase + r) * HS2 + n0] = fmaxf(c2[r] + bias, 0.f);
  }
  __syncthreads();

  // ---------------- Layer 3: (16 x 16) @ (16 x 8 padded to 16) ------------
  v8f c3 = {};
#pragma unroll
  for (int kk = 0; kk < 4; ++kk) {
    const int k0 = kk * 4 + koff;
    v2f av; av.x = h2[m * HS2 + k0]; av.y = h2[m * HS2 + k0 + 1];
    v2f bv; bv.x = 0.f; bv.y = 0.f;
    if (n0 < 8) { bv.x = w3[n0 * 16 + k0]; bv.y = w3[n0 * 16 + k0 + 1]; }
    c3 = wmma4(av, bv, c3);
  }

  // Masked sum over the tile's 16 patches, fold lane halves, atomic-add.
  float partial = 0.f;
#pragma unroll
  for (int r = 0; r < 8; ++r) {
    const int pi = t * 16 + mbase + r;
    partial += (pi < NP) ? c3[r] : 0.f;
  }
  partial += __shfl_xor(partial, 16, 32);    // combine M=0..7 and M=8..15
  if (activeWave && lane < 8)
    atomicAdd(&acc[b * 8 + lane], partial);  // global_atomic_add_f32
}

__global__ void zero_kernel(float* p, int n) {
  int i = blockIdx.x * blockDim.x + threadIdx.x;
  if (i < n) p[i] = 0.f;
}

__global__ void finalize_kernel(const float* __restrict__ acc,
                                const float* __restrict__ b3_0,
                                const float* __restrict__ b3_1,
                                const float* __restrict__ b3_2,
                                const float* __restrict__ b3_3,
                                const float* __restrict__ b3_4,
                                float* __restrict__ out, int B) {
  int i = blockIdx.x * blockDim.x + threadIdx.x;
  if (i >= B * 40) return;
  const int b = i / 40;
  const int c = i - b * 40;
  const int sc = c >> 3;
  const int n = c & 7;
  const float inv[5] = {1.f / 49284.f, 1.f / 12100.f, 1.f / 5329.f,
                        1.f / 2916.f, 1.f / 1849.f};
  const float* b3 = (sc == 0) ? b3_0
                  : (sc == 1) ? b3_1
                  : (sc == 2) ? b3_2
                  : (sc == 3) ? b3_3 : b3_4;
  out[i] = acc[(sc * B + b) * 8 + n] * inv[sc] + b3[n];
}

template <int P, int S, int WO, int NP>
static void launch_scale(const float* x, void* const* dp, float* acc, int B,
                         hipStream_t stream) {
  const int waves = B * ((NP + 15) / 16);
  const int blocks = (waves + 7) / 8;
  mlp_scale_kernel<P, S, WO, NP><<<blocks, 256, 0, stream>>>(
      x, (const float*)dp[0], (const float*)dp[1], (const float*)dp[2],
      (const float*)dp[3], (const float*)dp[4], acc, B);
}

extern "C" void kernel_launch(void* const* d_in, const int* in_sizes, int n_in,
                              void* d_out, int out_size, void* d_ws,
                              size_t ws_size, hipStream_t stream) {
  // d_in[0] = x (B,1,224,224) f32; then per scale: w1,b1,w2,b2,w3,b3
  const float* x = (const float*)d_in[0];
  const int B = in_sizes[0] / (224 * 224);
  float* acc = (float*)d_ws;                 // 5 * B * 8 floats
  const int accN = 5 * B * 8;
  zero_kernel<<<(accN + 255) / 256, 256, 0, stream>>>(acc, accN);

  launch_scale<3, 1, 222, 49284>(x, d_in + 1,  acc + 0 * B * 8, B, stream);
  launch_scale<5, 2, 110, 12100>(x, d_in + 7,  acc + 1 * B * 8, B, stream);
  launch_scale<7, 3, 73, 5329>  (x, d_in + 13, acc + 2 * B * 8, B, stream);
  launch_scale<9, 4, 54, 2916>  (x, d_in + 19, acc + 3 * B * 8, B, stream);
  launch_scale<11, 5, 43, 1849> (x, d_in + 25, acc + 4 * B * 8, B, stream);

  const int outN = B * 40;
  finalize_kernel<<<(outN + 255) / 256, 256, 0, stream>>>(
      acc, (const float*)d_in[6], (const float*)d_in[12],
      (const float*)d_in[18], (const float*)d_in[24], (const float*)d_in[30],
      (float*)d_out, B);
}